// AutoEncoder_40931038331116
// MI455X (gfx1250) — compile-verified
//
#include <hip/hip_runtime.h>
#include <cstdint>

#define B_TOTAL 65536
#define T_STEPS 28
#define IN_DIM  28
#define H_ENC_  4
#define H_DEC_  28
#define N_CLS_  10
#define GATE_D  112   // 4*H_DEC

typedef float v2f __attribute__((ext_vector_type(2)));
typedef float v8f __attribute__((ext_vector_type(8)));

static __device__ __forceinline__ v8f wmma4(v2f a, v2f b, v8f c) {
  // D(16x16,f32) = A(16x4,f32) * B(4x16,f32) + C
  return __builtin_amdgcn_wmma_f32_16x16x4_f32(false, a, false, b, (short)0, c,
                                               false, false);
}

// ---- branchless fast activations (native trans ops, no EXEC-mask paths) ----
static __device__ __forceinline__ float fexp(float x) {
  return __builtin_amdgcn_exp2f(x * 1.4426950408889634f);   // v_exp_f32
}
static __device__ __forceinline__ float sigmf(float x) {
  return __builtin_amdgcn_rcpf(1.0f + fexp(-x));            // v_rcp_f32
}
static __device__ __forceinline__ float tanh_fast(float x) {
#if __has_builtin(__builtin_amdgcn_tanhf)
  return __builtin_amdgcn_tanhf(x);                         // v_tanh_f32
#else
  return 1.0f - 2.0f * __builtin_amdgcn_rcpf(1.0f + fexp(2.0f * x));
#endif
}

__global__ __launch_bounds__(128) void lstm_ae_fused(
    const float* __restrict__ x,
    const float* __restrict__ eWih, const float* __restrict__ eWhh,
    const float* __restrict__ eb,
    const float* __restrict__ dWih, const float* __restrict__ dWhh,
    const float* __restrict__ db,
    const float* __restrict__ UW, const float* __restrict__ Ub,
    float* __restrict__ out, float* __restrict__ pred)
{
  // LDS: [dec_Whh weights: 3136 floats, block-shared] + 4 wave-private slabs.
  __shared__ float smem[3136 + 4 * 2304];
  float* wWhh = smem;

  const int tid  = threadIdx.x;
  const int lane = tid & 31;
  const int wave = tid >> 5;
  const int col  = lane & 15;              // B/C/D column owned by this lane
  const int koff = (lane >> 4) << 1;       // K-pair offset for A/B fragments
  const int rowA = lane & 15;              // A-matrix row owned by this lane
  const int rowC = (lane < 16) ? 0 : 8;    // first C/D row in this lane
  const int b0   = (blockIdx.x * 4 + wave) * 16;

  float* gates = smem + 3136 + wave * 2304;  // [16][112] (reused as 16x16)
  float* hEnc  = gates + 1792;               // [16][4]
  float* hDec  = hEnc + 64;                  // [16][28]

  // cooperative, coalesced copy of decoder recurrence weights into LDS
  for (int i = tid; i < GATE_D * H_DEC_; i += 128) wWhh[i] = dWhh[i];

  // ------------- small loop-invariant weight fragments in VGPRs -------------
  v2f BxE[7];
#pragma unroll
  for (int kk = 0; kk < 7; ++kk)
    BxE[kk] = *(const v2f*)(eWih + col * IN_DIM + 4 * kk + koff);
  v2f  BhE   = *(const v2f*)(eWhh + col * H_ENC_ + koff);
  float biasE = eb[col];

  v2f BxD[7];
  float biasD[7];
#pragma unroll
  for (int nt = 0; nt < 7; ++nt) {
    const int ng = nt * 16 + col;
    BxD[nt]   = *(const v2f*)(dWih + ng * H_ENC_ + koff);
    biasD[nt] = db[ng];
  }

  // ---------------- zero-init recurrent state ----------------
  hEnc[lane] = 0.0f;
  hEnc[lane + 32] = 0.0f;
#pragma unroll
  for (int r = 0; r < 14; ++r) hDec[lane + 32 * r] = 0.0f;

  float cE0 = 0.0f, cE1 = 0.0f;   // encoder cell state (2 elems/lane)
  float cD[14];                   // decoder cell state (14 elems/lane)
#pragma unroll
  for (int r = 0; r < 14; ++r) cD[r] = 0.0f;

  __syncthreads();                // weights visible to all waves

  const float* xr = x + (size_t)(b0 + rowA) * (T_STEPS * IN_DIM);
  const int jE = lane & 3;
  const int mE = lane >> 2;

  // ============================ time loop ============================
  for (int t = 0; t < T_STEPS; ++t) {
    if (t + 1 < T_STEPS)
      __builtin_prefetch(xr + (t + 1) * IN_DIM, 0, 1);

    // -------- encoder LSTM cell: gates[16x16] --------
    v8f acc;
#pragma unroll
    for (int v = 0; v < 8; ++v) acc[v] = biasE;
#pragma unroll
    for (int kk = 0; kk < 7; ++kk) {
      v2f a = *(const v2f*)(xr + t * IN_DIM + 4 * kk + koff);
      acc = wmma4(a, BxE[kk], acc);
    }
    {
      v2f ah = *(const v2f*)(hEnc + rowA * H_ENC_ + koff);
      acc = wmma4(ah, BhE, acc);
    }
#pragma unroll
    for (int v = 0; v < 8; ++v) gates[(rowC + v) * 16 + col] = acc[v];

    // encoder nonlinearity (wave-internal LDS transpose; DS ops in-order)
    {
      float i0 = gates[mE * 16 + jE],      f0 = gates[mE * 16 + 4 + jE];
      float g0 = gates[mE * 16 + 8 + jE],  o0 = gates[mE * 16 + 12 + jE];
      cE0 = sigmf(f0) * cE0 + sigmf(i0) * tanh_fast(g0);
      float h0 = sigmf(o0) * tanh_fast(cE0);
      const int m1 = mE + 8;
      float i1 = gates[m1 * 16 + jE],      f1 = gates[m1 * 16 + 4 + jE];
      float g1 = gates[m1 * 16 + 8 + jE],  o1 = gates[m1 * 16 + 12 + jE];
      cE1 = sigmf(f1) * cE1 + sigmf(i1) * tanh_fast(g1);
      float h1 = sigmf(o1) * tanh_fast(cE1);
      hEnc[mE * H_ENC_ + jE] = h0;
      hEnc[m1 * H_ENC_ + jE] = h1;
    }

    // -------- decoder LSTM cell: gates[16x112], input z_t == hEnc --------
    v2f az = *(const v2f*)(hEnc + rowA * H_ENC_ + koff);
#pragma unroll
    for (int nt = 0; nt < 7; ++nt) {
      const int ng = nt * 16 + col;
      v8f d;
#pragma unroll
      for (int v = 0; v < 8; ++v) d[v] = biasD[nt];
      d = wmma4(az, BxD[nt], d);
#pragma unroll
      for (int kk = 0; kk < 7; ++kk) {
        v2f ah = *(const v2f*)(hDec + rowA * H_DEC_ + 4 * kk + koff);
        v2f bh = *(const v2f*)(wWhh + ng * H_DEC_ + 4 * kk + koff);
        d = wmma4(ah, bh, d);
      }
#pragma unroll
      for (int v = 0; v < 8; ++v)
        gates[(rowC + v) * GATE_D + nt * 16 + col] = d[v];
    }

    // decoder nonlinearity + stream h to global out[b, t, :]
#pragma unroll
    for (int r = 0; r < 14; ++r) {
      const int e = lane + 32 * r;
      const int m = e / 28;
      const int j = e - m * 28;
      float ig = gates[m * GATE_D + j];
      float fg = gates[m * GATE_D + 28 + j];
      float gg = gates[m * GATE_D + 56 + j];
      float og = gates[m * GATE_D + 84 + j];
      float c  = sigmf(fg) * cD[r] + sigmf(ig) * tanh_fast(gg);
      cD[r] = c;
      float h = sigmf(og) * tanh_fast(c);
      hDec[m * H_DEC_ + j] = h;
      out[(size_t)(b0 + m) * (T_STEPS * H_DEC_) + t * H_DEC_ + j] = h;
    }
  }

  // ================= classifier: logits = h @ U_W^T + U_b =================
  v2f BU[7];
  float biasU = (col < N_CLS_) ? Ub[col] : 0.0f;
#pragma unroll
  for (int kk = 0; kk < 7; ++kk) {
    if (col < N_CLS_)
      BU[kk] = *(const v2f*)(UW + col * H_DEC_ + 4 * kk + koff);
    else {
      BU[kk].x = 0.0f;
      BU[kk].y = 0.0f;
    }
  }
  v8f lg;
#pragma unroll
  for (int v = 0; v < 8; ++v) lg[v] = biasU;
#pragma unroll
  for (int kk = 0; kk < 7; ++kk) {
    v2f ah = *(const v2f*)(hDec + rowA * H_DEC_ + 4 * kk + koff);
    lg = wmma4(ah, BU[kk], lg);
  }
#pragma unroll
  for (int v = 0; v < 8; ++v) gates[(rowC + v) * 16 + col] = lg[v];

  // row softmax over 10 classes (lanes 0..15, one batch row each)
  if (lane < 16) {
    const int m = lane;
    float l[N_CLS_];
    float mx = -1e30f;
#pragma unroll
    for (int j = 0; j < N_CLS_; ++j) {
      l[j] = gates[m * 16 + j];
      mx = fmaxf(mx, l[j]);
    }
    float s = 0.0f;
#pragma unroll
    for (int j = 0; j < N_CLS_; ++j) {
      l[j] = fexp(l[j] - mx);
      s += l[j];
    }
    const float inv = __builtin_amdgcn_rcpf(s);
#pragma unroll
    for (int j = 0; j < N_CLS_; ++j)
      pred[(size_t)(b0 + m) * N_CLS_ + j] = l[j] * inv;
  }
}

extern "C" void kernel_launch(void* const* d_in, const int* in_sizes, int n_in,
                              void* d_out, int out_size, void* d_ws, size_t ws_size,
                              hipStream_t stream) {
  (void)in_sizes; (void)n_in; (void)out_size; (void)d_ws; (void)ws_size;
  const float* x    = (const float*)d_in[0];
  const float* eWih = (const float*)d_in[1];
  const float* eWhh = (const float*)d_in[2];
  const float* eb   = (const float*)d_in[3];
  const float* dWih = (const float*)d_in[4];
  const float* dWhh = (const float*)d_in[5];
  const float* db   = (const float*)d_in[6];
  const float* UW   = (const float*)d_in[7];
  const float* Ub   = (const float*)d_in[8];

  float* out  = (float*)d_out;
  float* pred = out + (size_t)B_TOTAL * T_STEPS * H_DEC_;

  dim3 grid(B_TOTAL / 64);   // 4 waves/block * 16 rows/wave
  lstm_ae_fused<<<grid, 128, 0, stream>>>(x, eWih, eWhh, eb, dWih, dWhh, db,
                                          UW, Ub, out, pred);
}